// postprocess_23210003268257
// MI455X (gfx1250) — compile-verified
//
#include <hip/hip_runtime.h>

// ---------------------------------------------------------------------------
// Problem constants
// ---------------------------------------------------------------------------
#define BATCH 16
#define NCLS  9
#define HH    264
#define WW    480
#define NPIX  (HH * WW)          // 126720
#define KTOP  50
#define NDET  (BATCH * KTOP)     // 800
#define NREGC 46
#define PI_F     3.14159265358979323846f
#define TWOPI_F  6.28318530717958647692f

#define TROWS 16                 // output rows per tile
#define LROWS (TROWS + 4)        // staged rows (halo +-2)
#define TILE_ELEMS (LROWS * WW)  // 9600 floats = 38400 B

__constant__ float c_dim_mean[9][3] = {
    {3.99331126f, 1.54370861f, 1.64175497f},
    {0.295f, 1.6f, 0.3175f},
    {1.34645161f, 1.55322581f, 0.3883871f},
    {2.503f, 1.72f, 1.077f},
    {9.1775f, 2.95f, 2.3425f},
    {10.3655102f, 3.31632653f, 2.45469388f},
    {6.016911083f, 3.412001685f, 2.2783185f},
    {4.824963f, 2.046904f, 1.78939f},
    {8.8040879f, 2.916193f, 2.07649252f}};

__constant__ float c_alpha_c[4] = {0.0f, 1.57079632679489662f,
                                   3.14159265358979323846f,
                                   -1.57079632679489662f};

// ---------------------------------------------------------------------------
// gfx1250 data movers, probe-guarded with graceful fallbacks:
//   1) TDM tensor_load_to_lds (scalar 2D DMA, TENSORcnt) — 6-arg clang-23 form
//   2) per-lane global_load_async_to_lds_b128 (ASYNCcnt)
//   3) plain LDS stores
// ---------------------------------------------------------------------------
#if defined(__has_builtin)
#if __has_builtin(__builtin_amdgcn_tensor_load_to_lds) && \
    __has_builtin(__builtin_amdgcn_s_wait_tensorcnt)
#define HAVE_TDM 1
#endif
#if __has_builtin(__builtin_amdgcn_global_load_async_to_lds_b128)
#define HAVE_ASYNC_LDS 1
#endif
#endif

typedef unsigned int u32x4 __attribute__((ext_vector_type(4)));
typedef int i32x8 __attribute__((ext_vector_type(8)));
typedef int i32x4 __attribute__((ext_vector_type(4)));

typedef int v4i_vs __attribute__((vector_size(16)));
typedef __attribute__((address_space(1))) v4i_vs* gptr128_t;
typedef __attribute__((address_space(3))) v4i_vs* lptr128_t;

__device__ __forceinline__ void lds_copy16(void* ldsDst, const void* gSrc) {
#ifdef HAVE_ASYNC_LDS
  __builtin_amdgcn_global_load_async_to_lds_b128(
      (gptr128_t)(unsigned long long)gSrc,
      (lptr128_t)(unsigned)(unsigned long long)ldsDst, 0, 0);
#else
  *(float4*)ldsDst = *(const float4*)gSrc;
#endif
}

__device__ __forceinline__ void lds_copy_wait() {
#ifdef HAVE_ASYNC_LDS
#if __has_builtin(__builtin_amdgcn_s_wait_asynccnt)
  __builtin_amdgcn_s_wait_asynccnt(0);
#else
  asm volatile("s_wait_asynccnt 0" ::: "memory");
#endif
#endif
}

// ---------------------------------------------------------------------------
// In-LDS bitonic sort, descending, blockDim.x == N, N power of two
// ---------------------------------------------------------------------------
template <int N>
__device__ __forceinline__ void bitonic_desc(unsigned long long* key) {
  const int tid = threadIdx.x;
  for (int k = 2; k <= N; k <<= 1) {
    for (int j = k >> 1; j > 0; j >>= 1) {
      int ixj = tid ^ j;
      if (ixj > tid) {
        unsigned long long a = key[tid], b = key[ixj];
        bool sw = ((tid & k) == 0) ? (a < b) : (a > b);
        if (sw) { key[tid] = b; key[ixj] = a; }
      }
      __syncthreads();
    }
  }
}

// ---------------------------------------------------------------------------
// Kernel 1: per-(b,c) NMS + exact top-50 via radix-select
// grid = 144 blocks, block = 1024 threads
// ---------------------------------------------------------------------------
__global__ __launch_bounds__(1024) void topk_class_kernel(
    const float* __restrict__ heat, unsigned* __restrict__ outScore,
    unsigned* __restrict__ outInd) {
  const int bc = blockIdx.x;
  const int tid = threadIdx.x;
  const float* base = heat + (size_t)bc * NPIX;

  __shared__ __align__(16) float tile[TILE_ELEMS];
  __shared__ unsigned hist[256];
  __shared__ unsigned s_sel, s_need, s_cnt;
  __shared__ unsigned long long cand[1024];

  if (tid == 0) s_need = KTOP;
  unsigned prefix = 0, knownMask = 0;
  __syncthreads();

  // ---- stage rows [ty-2, ty+LROWS-3] into LDS ----
  // TDM path: one 2D descriptor per tile; bottom halo rows past the image
  // are zero-filled by TDM OOB semantics (tensor_dim1 = HH - rowsLo); only
  // the two above-image rows of the first tile need manual zeros.
  auto load_tile = [&](int ty) {
#ifdef HAVE_TDM
    const int topMiss = (ty == 0) ? 2 : 0;
    const int rowsLo = ty - 2 + topMiss;
    const int nrows = LROWS - topMiss;
    if (topMiss) {
      for (int e = tid; e < 2 * WW; e += 1024) tile[e] = 0.f;
    }
    if (tid < 32) {  // single wave issues the DMA
      unsigned ldsAddr =
          (unsigned)(unsigned long long)(void*)&tile[topMiss * WW];
      unsigned long long gAddr =
          (unsigned long long)(const void*)(base + (size_t)rowsLo * WW);
      u32x4 g0;
      g0.x = 1u;                                   // count=1, gather off
      g0.y = ldsAddr;                              // lds_addr
      g0.z = (unsigned)(gAddr & 0xFFFFFFFFu);      // global_addr[31:0]
      g0.w = (unsigned)((gAddr >> 32) & 0x1FFFFFFu) | (2u << 30);  // type=2
      i32x8 g1;
      g1[0] = 0x00020000;                          // data_size = 4B
      g1[1] = (int)((unsigned)WW << 16);           // tensor_dim0[15:0]
      g1[2] = (int)(((unsigned)(HH - rowsLo) & 0xFFFFu) << 16);  // dim1 lo
      g1[3] = (int)((unsigned)WW << 16);           // tile_dim0
      g1[4] = nrows;                               // tile_dim1 (tile_dim2=0)
      g1[5] = WW;                                  // tensor_dim0_stride lo32
      g1[6] = 0;
      g1[7] = 0;
      i32x4 gz4 = {0, 0, 0, 0};                    // 2D: groups 2/3 unused
      i32x8 gz8 = {0, 0, 0, 0, 0, 0, 0, 0};
      __builtin_amdgcn_tensor_load_to_lds(g0, g1, gz4, gz4, gz8, 0);
      __builtin_amdgcn_s_wait_tensorcnt(0);
    }
    __syncthreads();
#else
    for (int q = tid; q < TILE_ELEMS / 4; q += 1024) {
      int elem = q * 4;
      int slot = elem / WW;
      int col = elem % WW;
      int gy = ty - 2 + slot;
      if (gy >= 0 && gy < HH) {
        lds_copy16(&tile[elem], base + (size_t)gy * WW + col);
      } else {
        tile[elem] = 0.f; tile[elem + 1] = 0.f;
        tile[elem + 2] = 0.f; tile[elem + 3] = 0.f;
      }
    }
    lds_copy_wait();
    __syncthreads();
#endif
  };

  // ---- NMS value as sortable uint for local pixel ----
  auto nms_u = [&](int ly, int x) -> unsigned {
    float v = tile[(ly + 2) * WW + x];
    int c0 = x - 2; if (c0 < 0) c0 = 0;
    int c1 = x + 2; if (c1 > WW - 1) c1 = WW - 1;
    float m = v;
    #pragma unroll
    for (int r = 0; r < 5; ++r) {
      const float* row = &tile[(ly + r) * WW];
      for (int c = c0; c <= c1; ++c) m = fmaxf(m, row[c]);
    }
    return (m <= v) ? __float_as_uint(v) : 0u;
  };

  // ---- 4 radix-select passes (MSB -> LSB byte) ----
  for (int pass = 0; pass < 4; ++pass) {
    const int shift = 24 - 8 * pass;
    if (tid < 256) hist[tid] = 0u;
    __syncthreads();

    for (int ty = 0; ty < HH; ty += TROWS) {
      load_tile(ty);
      int rows = HH - ty; if (rows > TROWS) rows = TROWS;
      for (int p = tid; p < rows * WW; p += 1024) {
        int ly = p / WW, x = p % WW;
        unsigned u = nms_u(ly, x);
        if ((u & knownMask) == prefix)
          atomicAdd(&hist[(u >> shift) & 255u], 1u);
      }
      __syncthreads();  // tile reuse barrier
    }

    if (tid == 0) {
      unsigned need = s_need, cum = 0; int sel = 0;
      for (int bn = 255; bn >= 0; --bn) {
        unsigned c = hist[bn];
        if (cum + c >= need) { sel = bn; break; }
        cum += c;
      }
      s_sel = (unsigned)sel;
      s_need = need - cum;
    }
    __syncthreads();
    prefix |= (s_sel << shift);
    knownMask |= (0xFFu << shift);
    __syncthreads();
  }
  const unsigned T = prefix;  // exact K-th largest sortable value

  // ---- collect candidates >= T ----
  if (tid == 0) s_cnt = 0u;
  __syncthreads();
  for (int ty = 0; ty < HH; ty += TROWS) {
    load_tile(ty);
    int rows = HH - ty; if (rows > TROWS) rows = TROWS;
    for (int p = tid; p < rows * WW; p += 1024) {
      int ly = p / WW, x = p % WW;
      unsigned u = nms_u(ly, x);
      if (u >= T && u != 0u) {
        unsigned pos = atomicAdd(&s_cnt, 1u);
        unsigned idx = (unsigned)((ty + ly) * WW + x);
        if (pos < 1024u)
          cand[pos] = ((unsigned long long)u << 32) | (0xFFFFFFFFu - idx);
      }
    }
    __syncthreads();
  }
  unsigned cnt = s_cnt; if (cnt > 1024u) cnt = 1024u;
  if ((unsigned)tid >= cnt) cand[tid] = 0ull;   // pad (blockDim == 1024)
  __syncthreads();

  bitonic_desc<1024>(cand);

  if (tid < KTOP) {
    unsigned long long kk = cand[tid];
    outScore[bc * KTOP + tid] = (unsigned)(kk >> 32);
    outInd[bc * KTOP + tid] = 0xFFFFFFFFu - (unsigned)(kk & 0xFFFFFFFFu);
  }
}

// ---------------------------------------------------------------------------
// Kernel 2: per-batch merge of 9*50 candidates -> top 50
// grid = 16 blocks, block = 512 threads
// ---------------------------------------------------------------------------
__global__ __launch_bounds__(512) void topk_merge_kernel(
    const unsigned* __restrict__ scoreU, const unsigned* __restrict__ indU,
    float* __restrict__ detScore, int* __restrict__ detInd,
    int* __restrict__ detCls) {
  const int b = blockIdx.x;
  const int tid = threadIdx.x;
  __shared__ unsigned long long key[512];

  if (tid < NCLS * KTOP) {
    int c = tid / KTOP, j = tid % KTOP;
    unsigned u = scoreU[(b * NCLS + c) * KTOP + j];
    key[tid] = ((unsigned long long)u << 32) | (0xFFFFFFFFu - (unsigned)tid);
  } else {
    key[tid] = 0ull;
  }
  __syncthreads();

  bitonic_desc<512>(key);

  if (tid < KTOP) {
    unsigned long long kk = key[tid];
    unsigned u = (unsigned)(kk >> 32);
    unsigned t = 0xFFFFFFFFu - (unsigned)(kk & 0xFFFFFFFFu);
    int o = b * KTOP + tid;
    if (t < (unsigned)(NCLS * KTOP)) {
      int c = (int)(t / KTOP), j = (int)(t % KTOP);
      unsigned ind = indU[(b * NCLS + c) * KTOP + j];
      if (ind >= (unsigned)NPIX) ind = 0u;  // pad safety
      detScore[o] = __uint_as_float(u);
      detInd[o] = (int)ind;
      detCls[o] = c;
    } else {
      detScore[o] = 0.f; detInd[o] = 0; detCls[o] = 0;
    }
  }
}

// ---------------------------------------------------------------------------
// Kernel 3: gather regs + geometry + segment-max dedup. 1 block, 1024 thr.
// ---------------------------------------------------------------------------
__global__ __launch_bounds__(1024) void finalize_kernel(
    const float* __restrict__ regs, const float* __restrict__ calib,
    const float* __restrict__ detScore, const int* __restrict__ detInd,
    const int* __restrict__ detCls, float* __restrict__ out) {
  __shared__ float sKey[NDET];
  __shared__ int sInd[NDET];
  const int i = threadIdx.x;

  float score = 0.f; int ind = 0, cls = 0; bool th = false;
  if (i < NDET) {
    score = detScore[i];
    ind = detInd[i]; if (ind < 0) ind = 0; if (ind >= NPIX) ind = NPIX - 1;
    cls = detCls[i];
    th = (score >= 0.29f);
    sKey[i] = th ? ((float)cls * 10000.0f - (float)i) : -__builtin_inff();
    sInd[i] = ind;
  }
  __syncthreads();
  if (i >= NDET) return;

  bool valid = th;
  if (th) {
    float myKey = sKey[i];
    for (int j = 0; j < NDET; ++j)
      if (sInd[j] == ind && sKey[j] > myKey) { valid = false; break; }
  }

  const int b = i / KTOP;
  const float* pbase = regs + (size_t)b * NREGC * NPIX + (size_t)ind;
  auto P = [&](int ch) -> float { return pbase[(size_t)ch * NPIX]; };

  float r0 = fmaxf(P(0), 0.f), r1 = fmaxf(P(1), 0.f);
  float r2 = fmaxf(P(2), 0.f), r3 = fmaxf(P(3), 0.f);
  float ox = P(4), oy = P(5);
  float d0 = P(22), d1 = P(23), d2 = P(24);
  float o[16];
  #pragma unroll
  for (int k = 0; k < 16; ++k) o[k] = P(25 + k);
  float dep_off = P(41);

  float xs = (float)(ind % WW), ys = (float)(ind / WW);
  float cx = xs + ox, cy = ys + oy;

  float bx0 = fminf(fmaxf((cx - r0) * 4.f, 0.f), 1920.f);
  float by0 = fminf(fmaxf((cy - r1) * 4.f, 0.f), 1056.f);
  float bx1 = fminf(fmaxf((cx + r2) * 4.f, 0.f), 1920.f);
  float by1 = fminf(fmaxf((cy + r3) * 4.f, 0.f), 1056.f);

  float dm0 = expf(d0) * c_dim_mean[cls][0];
  float dm1 = expf(d1) * c_dim_mean[cls][1];
  float dm2 = expf(d2) * c_dim_mean[cls][2];

  float sig = 1.f / (1.f + expf(-dep_off));
  float depth = fminf(fmaxf(1.f / sig - 1.f, 0.1f), 200.f);

  float fu = calib[0], cu = calib[2], fv = calib[5], cv = calib[6];
  float bx = calib[3] / (-fu), by = calib[7] / (-fv);
  float px = cx * 4.f, py = cy * 4.f;
  float locx = (px - cu) * depth / fu + bx;
  float locy = (py - cv) * depth / fv + by;

  int bsel = 0;
  float best = o[1] - o[0];
  #pragma unroll
  for (int k = 1; k < 4; ++k) {
    float dd = o[2 * k + 1] - o[2 * k];
    if (dd > best) { best = dd; bsel = k; }
  }
  float s0 = o[8 + 2 * bsel], s1 = o[8 + 2 * bsel + 1];
  float alpha = atanf(s0 / s1) + c_alpha_c[bsel];
  float ray = atanf(locx / depth);
  float roty = alpha + ray;
  if (roty > PI_F) roty -= TWOPI_F;
  if (roty < -PI_F) roty += TWOPI_F;
  if (alpha > PI_F) alpha -= TWOPI_F;
  if (alpha < -PI_F) alpha += TWOPI_F;

  float* row = out + (size_t)i * 14;
  row[0] = bx0; row[1] = by0; row[2] = bx1; row[3] = by1;
  row[4] = dm0; row[5] = dm1; row[6] = dm2;
  row[7] = depth;
  row[8] = locx; row[9] = locy; row[10] = depth;
  row[11] = roty; row[12] = alpha; row[13] = score;

  out[NDET * 14 + i] = valid ? 1.0f : 0.0f;   // valid (bool) as float
  out[NDET * 15 + i] = (float)cls;            // cls_id as float
}

// ---------------------------------------------------------------------------
// Launch
// ---------------------------------------------------------------------------
extern "C" void kernel_launch(void* const* d_in, const int* in_sizes, int n_in,
                              void* d_out, int out_size, void* d_ws,
                              size_t ws_size, hipStream_t stream) {
  (void)in_sizes; (void)n_in; (void)out_size; (void)ws_size;
  const float* heat = (const float*)d_in[0];   // (16,9,264,480) f32
  const float* regs = (const float*)d_in[1];   // (16,46,264,480) f32
  const float* calib = (const float*)d_in[2];  // (3,4) f32
  float* out = (float*)d_out;                  // 800*14 + 800 + 800 f32

  unsigned* scoreU = (unsigned*)d_ws;                       // 144*50
  unsigned* indU = scoreU + BATCH * NCLS * KTOP;            // 144*50
  float* detScore = (float*)(indU + BATCH * NCLS * KTOP);   // 800
  int* detInd = (int*)(detScore + NDET);                    // 800
  int* detCls = detInd + NDET;                              // 800

  topk_class_kernel<<<BATCH * NCLS, 1024, 0, stream>>>(heat, scoreU, indU);
  topk_merge_kernel<<<BATCH, 512, 0, stream>>>(scoreU, indU, detScore, detInd,
                                               detCls);
  finalize_kernel<<<1, 1024, 0, stream>>>(regs, calib, detScore, detInd,
                                          detCls, out);
}